// DLRM_Net_81724637708434
// MI455X (gfx1250) — compile-verified
//
#include <hip/hip_runtime.h>
#include <math.h>

// ---------------------------------------------------------------------------
// DLRM forward for MI455X (gfx1250), fp32 via V_WMMA_F32_16X16X4_F32.
// ---------------------------------------------------------------------------

typedef __attribute__((ext_vector_type(2))) float v2f;
typedef __attribute__((ext_vector_type(8))) float v8f;

#define WMMA_F32(a, b, c) \
  __builtin_amdgcn_wmma_f32_16x16x4_f32(false, (a), false, (b), (short)0, (c), false, false)

#define BATCH 4096
#define NUM_TABLES 26
#define VOCAB 100000
#define EMB_D 64
#define NROWS 27           // 1 (dense x) + 26 tables
#define RDIM 415           // 64 + 351 lower-triangle pairs

// ---------------------------------------------------------------------------
// Generic  C[M,N] = act(A[M,K] @ W[N,K]^T + bias[N])
// One wave -> 16 (M) x 64 (N) output strip. Requires M%16==0, N%64==0, any K.
//
// Fragment layout (wave32, fp32 WMMA 16x16x4):
//   A 16x4 : lane<16 holds M=lane,   K=k0+0/k0+1 ; lane>=16 holds M=lane-16, K=k0+2/k0+3
//   B 4x16 : lane<16 holds N=lane,   K=k0+0/k0+1 ; lane>=16 holds N=lane-16, K=k0+2/k0+3
//   D 16x16: lane<16 VGPR j -> (M=j,   N=lane) ; lane>=16 VGPR j -> (M=8+j, N=lane-16)
// Both A and B fragments are contiguous float2 loads along the K-major rows.
// ---------------------------------------------------------------------------
__global__ __launch_bounds__(256) void gemm_bias_relu_wmma(
    const float* __restrict__ A, const float* __restrict__ W,
    const float* __restrict__ bias, float* __restrict__ C,
    int M, int N, int K, int doRelu)
{
  const int lane = threadIdx.x & 31;
  const int wave = blockIdx.x * (blockDim.x >> 5) + (threadIdx.x >> 5);
  const int tilesN = N >> 6;
  const int tilesM = M >> 4;
  if (wave >= tilesM * tilesN) return;              // wave-uniform branch (EXEC stays full)

  const int m0 = (wave / tilesN) << 4;
  const int n0 = (wave % tilesN) << 6;
  const int half = lane >> 4;
  const int r    = lane & 15;
  const int kb   = half * 2;                        // per-lane K sub-offset

  const float* __restrict__ Ar = A + (size_t)(m0 + r) * K;
  const float* __restrict__ W0 = W + (size_t)(n0 + r) * K;
  const float* __restrict__ W1 = W0 + (size_t)16 * K;
  const float* __restrict__ W2 = W0 + (size_t)32 * K;
  const float* __restrict__ W3 = W0 + (size_t)48 * K;

  v8f acc0 = {}, acc1 = {}, acc2 = {}, acc3 = {};

  const int K4 = K & ~3;
  for (int k0 = 0; k0 < K4; k0 += 4) {
    v2f a, b0, b1, b2, b3;
    a.x  = Ar[k0 + kb]; a.y  = Ar[k0 + kb + 1];
    b0.x = W0[k0 + kb]; b0.y = W0[k0 + kb + 1];
    b1.x = W1[k0 + kb]; b1.y = W1[k0 + kb + 1];
    b2.x = W2[k0 + kb]; b2.y = W2[k0 + kb + 1];
    b3.x = W3[k0 + kb]; b3.y = W3[k0 + kb + 1];
    acc0 = WMMA_F32(a, b0, acc0);
    acc1 = WMMA_F32(a, b1, acc1);
    acc2 = WMMA_F32(a, b2, acc2);
    acc3 = WMMA_F32(a, b3, acc3);
  }
  if (K4 < K) {                                     // guarded tail (K=13, K=415)
    const int k0 = K4;
    const bool g0 = (k0 + kb)     < K;
    const bool g1 = (k0 + kb + 1) < K;
    v2f a, b0, b1, b2, b3;
    a.x  = g0 ? Ar[k0 + kb] : 0.0f; a.y  = g1 ? Ar[k0 + kb + 1] : 0.0f;
    b0.x = g0 ? W0[k0 + kb] : 0.0f; b0.y = g1 ? W0[k0 + kb + 1] : 0.0f;
    b1.x = g0 ? W1[k0 + kb] : 0.0f; b1.y = g1 ? W1[k0 + kb + 1] : 0.0f;
    b2.x = g0 ? W2[k0 + kb] : 0.0f; b2.y = g1 ? W2[k0 + kb + 1] : 0.0f;
    b3.x = g0 ? W3[k0 + kb] : 0.0f; b3.y = g1 ? W3[k0 + kb + 1] : 0.0f;
    acc0 = WMMA_F32(a, b0, acc0);
    acc1 = WMMA_F32(a, b1, acc1);
    acc2 = WMMA_F32(a, b2, acc2);
    acc3 = WMMA_F32(a, b3, acc3);
  }

  // Epilogue: bias + relu + store per D-matrix layout.
  v8f accs[4] = {acc0, acc1, acc2, acc3};
#pragma unroll
  for (int t = 0; t < 4; ++t) {
    const int n = n0 + t * 16 + r;
    const float bn = bias[n];
#pragma unroll
    for (int j = 0; j < 8; ++j) {
      const int m = m0 + j + 8 * half;
      float v = accs[t][j] + bn;
      if (doRelu) v = fmaxf(v, 0.0f);
      C[(size_t)m * N + n] = v;
    }
  }
}

// ---------------------------------------------------------------------------
// Build T[b, 0:27, 0:64]: row 0 = bottom-MLP output x[b], rows 1..26 = embedding
// gather (each "bag" has exactly one index since lS_o == arange(B)).
// One thread moves one float4.
// ---------------------------------------------------------------------------
__global__ __launch_bounds__(256) void build_T_kernel(
    const float* __restrict__ x, const float* __restrict__ emb,
    const int* __restrict__ lSi, float* __restrict__ T)
{
  const int gid = blockIdx.x * blockDim.x + threadIdx.x;
  const int total = BATCH * NROWS * (EMB_D / 4);
  if (gid >= total) return;
  const int q   = gid & 15;                 // float4 slot within 64
  const int row = (gid >> 4) % NROWS;
  const int b   = gid / (NROWS * 16);

  float4 v;
  if (row == 0) {
    v = ((const float4*)(x + (size_t)b * EMB_D))[q];
  } else {
    const int t  = row - 1;
    const int ix = lSi[t * BATCH + b];
    v = ((const float4*)(emb + ((size_t)t * VOCAB + ix) * EMB_D))[q];
  }
  ((float4*)(T + ((size_t)b * NROWS + row) * EMB_D))[q] = v;
}

// ---------------------------------------------------------------------------
// Interaction: one wave per batch. Z = T_b @ T_b^T with T_b 27x64 (padded to
// 32). A-frag and B-frag of the same row-tile carry identical lane values
// under the symmetric WMMA layouts, so 2 loads feed 3 WMMAs per K-step.
// Emit R[b] = [ x(64) | Z[i,j] for i>j, p = i(i-1)/2 + j ].
// ---------------------------------------------------------------------------
__global__ __launch_bounds__(256) void interact_wmma(
    const float* __restrict__ T, float* __restrict__ R)
{
  const int lane = threadIdx.x & 31;
  const int b = blockIdx.x * (blockDim.x >> 5) + (threadIdx.x >> 5);
  const float* __restrict__ Tb = T + (size_t)b * NROWS * EMB_D;
  const int half = lane >> 4;
  const int r    = lane & 15;
  const int kb   = half * 2;
  const int row1 = 16 + r;
  const bool row1ok = row1 < NROWS;

  v8f c00 = {}, c10 = {}, c11 = {};
  for (int k0 = 0; k0 < EMB_D; k0 += 4) {
    v2f f0, f1;
    f0.x = Tb[r * EMB_D + k0 + kb];
    f0.y = Tb[r * EMB_D + k0 + kb + 1];
    f1.x = row1ok ? Tb[row1 * EMB_D + k0 + kb]     : 0.0f;   // zero-pad rows 27..31
    f1.y = row1ok ? Tb[row1 * EMB_D + k0 + kb + 1] : 0.0f;
    c00 = WMMA_F32(f0, f0, c00);      // tile (0,0): rows 0-15  x cols 0-15
    c10 = WMMA_F32(f1, f0, c10);      // tile (1,0): rows 16-31 x cols 0-15
    c11 = WMMA_F32(f1, f1, c11);      // tile (1,1): rows 16-31 x cols 16-31
  }

  float* __restrict__ Rb = R + (size_t)b * RDIM;
  // R[b, 0:64] = x[b] = T[b, 0, :]
  Rb[lane * 2]     = Tb[lane * 2];
  Rb[lane * 2 + 1] = Tb[lane * 2 + 1];

  auto scatter = [&](const v8f& c, int ti, int tj) {
    const int jc = tj + r;
#pragma unroll
    for (int j = 0; j < 8; ++j) {
      const int i = ti + j + 8 * half;
      if (i < NROWS && i > jc)
        Rb[64 + (i * (i - 1)) / 2 + jc] = c[j];
    }
  };
  scatter(c00, 0, 0);
  scatter(c10, 16, 0);
  scatter(c11, 16, 16);
}

// ---------------------------------------------------------------------------
// Final layer: out[b] = sigmoid(dot(z2[b,0:256], tw2[0,0:256]) + tb2)
// ---------------------------------------------------------------------------
__global__ __launch_bounds__(256) void top_final_kernel(
    const float* __restrict__ z, const float* __restrict__ w,
    const float* __restrict__ bias, float* __restrict__ out)
{
  const int b = blockIdx.x * blockDim.x + threadIdx.x;
  if (b >= BATCH) return;
  const float4* zb = (const float4*)(z + (size_t)b * 256);
  const float4* wv = (const float4*)w;
  float acc = bias[0];
#pragma unroll 8
  for (int q = 0; q < 64; ++q) {
    const float4 a = zb[q];
    const float4 c = wv[q];
    acc += a.x * c.x + a.y * c.y + a.z * c.z + a.w * c.w;
  }
  out[b] = 1.0f / (1.0f + expf(-acc));
}

// ---------------------------------------------------------------------------
extern "C" void kernel_launch(void* const* d_in, const int* in_sizes, int n_in,
                              void* d_out, int out_size, void* d_ws, size_t ws_size,
                              hipStream_t stream) {
  const float* dense_x = (const float*)d_in[0];
  /* d_in[1] = lS_o: offsets == arange(B) -> bag size 1, unused */
  const int*   lSi = (const int*)d_in[2];
  const float* emb = (const float*)d_in[3];
  const float* bw0 = (const float*)d_in[4];  const float* bb0 = (const float*)d_in[5];
  const float* bw1 = (const float*)d_in[6];  const float* bb1 = (const float*)d_in[7];
  const float* bw2 = (const float*)d_in[8];  const float* bb2 = (const float*)d_in[9];
  const float* tw0 = (const float*)d_in[10]; const float* tb0 = (const float*)d_in[11];
  const float* tw1 = (const float*)d_in[12]; const float* tb1 = (const float*)d_in[13];
  const float* tw2 = (const float*)d_in[14]; const float* tb2 = (const float*)d_in[15];
  float* out = (float*)d_out;

  // Workspace layout (floats); z1/z2 reuse h1/h2 (their consumers are done).
  float* ws = (float*)d_ws;
  float* h1 = ws;                                  // 4096*512
  float* h2 = h1 + (size_t)BATCH * 512;            // 4096*256
  float* xb = h2 + (size_t)BATCH * 256;            // 4096*64
  float* T  = xb + (size_t)BATCH * 64;             // 4096*27*64
  float* R  = T  + (size_t)BATCH * NROWS * EMB_D;  // 4096*415
  float* z1 = h1;
  float* z2 = h2;

  auto gemm = [&](const float* A, const float* W, const float* bias, float* C,
                  int M, int N, int K) {
    const int tiles  = (M / 16) * (N / 64);
    const int blocks = (tiles + 7) / 8;            // 8 waves / block
    gemm_bias_relu_wmma<<<blocks, 256, 0, stream>>>(A, W, bias, C, M, N, K, 1);
  };

  // Bottom MLP
  gemm(dense_x, bw0, bb0, h1, BATCH, 512, 13);
  gemm(h1,      bw1, bb1, h2, BATCH, 256, 512);
  gemm(h2,      bw2, bb2, xb, BATCH, 64,  256);

  // Embedding gather + T assembly
  {
    const int total  = BATCH * NROWS * (EMB_D / 4);
    build_T_kernel<<<(total + 255) / 256, 256, 0, stream>>>(xb, emb, lSi, T);
  }

  // Pairwise interaction -> R
  interact_wmma<<<BATCH / 8, 256, 0, stream>>>(T, R);

  // Top MLP
  gemm(R,  tw0, tb0, z1, BATCH, 512, RDIM);
  gemm(z1, tw1, tb1, z2, BATCH, 256, 512);
  top_final_kernel<<<BATCH / 256, 256, 0, stream>>>(z2, tw2, tb2, out);
}